// E90DualRate_20547123544803
// MI455X (gfx1250) — compile-verified
//
#include <hip/hip_runtime.h>

#define T_DIM 2048
#define B_DIM 8
#define H_DIM 16
#define KF 32
#define VF 64
#define KS 128
#define VS 64
#define CHUNK 16
#define NCHUNK (T_DIM / CHUNK)

typedef float v2f __attribute__((ext_vector_type(2)));
typedef float v8f __attribute__((ext_vector_type(8)));
typedef unsigned int v4u __attribute__((ext_vector_type(4)));
typedef int v4i __attribute__((ext_vector_type(4)));
typedef int v8i __attribute__((ext_vector_type(8)));

#define WMMA4(a, b, c) \
  __builtin_amdgcn_wmma_f32_16x16x4_f32(false, (a), false, (b), (short)0, (c), false, false)

// ---------------- Tensor Data Mover: 2D tile (nRows x rowElems f32) with
// hardware +1-DWORD row padding into LDS (D# per CDNA5 ISA ch.8) -------------
__device__ __forceinline__ void tdm_load_2d(unsigned lds_off, const float* gptr,
                                            unsigned rowElems, unsigned nRows,
                                            unsigned long long rowStrideElems,
                                            unsigned padIntervalCode) {
  unsigned long long ga = (unsigned long long)(size_t)gptr;
  v4u g0;
  g0.x = 1u;                                              // count=1, user descriptor
  g0.y = lds_off;                                         // LDS byte address
  g0.z = (unsigned)ga;                                    // global_addr[31:0]
  g0.w = (unsigned)((ga >> 32) & 0x1FFFFFFu) | (2u << 30);// global_addr[56:32], type=2
  v8i g1;
  // data_size=2 (4B), pad_enable=1, pad_interval=code (rowElems DWORDs), pad_amount=0 (1 DWORD)
  g1[0] = (int)((2u << 16) | (1u << 20) | (padIntervalCode << 22));
  g1[1] = (int)((rowElems & 0xFFFFu) << 16);              // abar=0 | tensor_dim0 lo
  g1[2] = (int)(rowElems >> 16);                          // tensor_dim0 hi | tensor_dim1 lo (=0)
  g1[3] = (int)(0x10u | (rowElems << 16));                // tensor_dim1 hi (dim1=1M) | tile_dim0
  g1[4] = (int)(nRows & 0xFFFFu);                         // tile_dim1 | tile_dim2=0
  g1[5] = (int)(unsigned)(rowStrideElems & 0xFFFFFFFFull);// tensor_dim0_stride lo
  g1[6] = (int)(unsigned)((rowStrideElems >> 32) & 0xFFFFull);
  g1[7] = 0;
  v4i z4 = {0, 0, 0, 0};                                  // 2D: groups 2/3 unused
  v8i z8 = {0, 0, 0, 0, 0, 0, 0, 0};
  __builtin_amdgcn_tensor_load_to_lds(g0, g1, z4, z4, z8, 0);
}

// ---------------- software-pipelined 16x16 WMMA-f32 GEMM helpers ------------
// C += A(row-major 16xKD) * B(row-major KDx16 slice)
template <int KD>
__device__ __forceinline__ v8f gemm_rr(const float* __restrict__ Ab, int lda,
                                       const float* __restrict__ Bb, int ldb,
                                       v8f acc, int lane16, int laneHi) {
  v2f a0, b0;
  a0.x = Ab[lane16 * lda + 2 * laneHi];
  a0.y = Ab[lane16 * lda + 2 * laneHi + 1];
  b0.x = Bb[(2 * laneHi) * ldb + lane16];
  b0.y = Bb[(2 * laneHi + 1) * ldb + lane16];
#pragma unroll
  for (int kb = 0; kb < KD; kb += 4) {
    v2f a1 = a0, b1 = b0;
    if (kb + 4 < KD) {                       // prefetch next fragment before WMMA
      a1.x = Ab[lane16 * lda + (kb + 4) + 2 * laneHi];
      a1.y = Ab[lane16 * lda + (kb + 4) + 2 * laneHi + 1];
      b1.x = Bb[((kb + 4) + 2 * laneHi) * ldb + lane16];
      b1.y = Bb[((kb + 4) + 2 * laneHi + 1) * ldb + lane16];
    }
    acc = WMMA4(a0, b0, acc);
    a0 = a1; b0 = b1;
  }
  return acc;
}

// C += A(row-major 16xKD) * Kt^T where B[k][n] = Kt[n][k] (Kt row-major 16xKD)
template <int KD>
__device__ __forceinline__ v8f gemm_rt(const float* __restrict__ Ab, int lda,
                                       const float* __restrict__ Kt, int ldk,
                                       v8f acc, int lane16, int laneHi) {
  v2f a0, b0;
  a0.x = Ab[lane16 * lda + 2 * laneHi];
  a0.y = Ab[lane16 * lda + 2 * laneHi + 1];
  b0.x = Kt[lane16 * ldk + 2 * laneHi];
  b0.y = Kt[lane16 * ldk + 2 * laneHi + 1];
#pragma unroll
  for (int kb = 0; kb < KD; kb += 4) {
    v2f a1 = a0, b1 = b0;
    if (kb + 4 < KD) {
      a1.x = Ab[lane16 * lda + (kb + 4) + 2 * laneHi];
      a1.y = Ab[lane16 * lda + (kb + 4) + 2 * laneHi + 1];
      b1.x = Kt[lane16 * ldk + (kb + 4) + 2 * laneHi];
      b1.y = Kt[lane16 * ldk + (kb + 4) + 2 * laneHi + 1];
    }
    acc = WMMA4(a0, b0, acc);
    a0 = a1; b0 = b1;
  }
  return acc;
}

// C += Kt^T * B : A[m][k] = Kt[k][m0+m] (time-major K staging), KD = CHUNK
__device__ __forceinline__ v8f gemm_tr(const float* __restrict__ Kt, int ldk, int m0,
                                       const float* __restrict__ Bb, int ldb,
                                       v8f acc, int lane16, int laneHi) {
  v2f a0, b0;
  a0.x = Kt[(2 * laneHi) * ldk + m0 + lane16];
  a0.y = Kt[(2 * laneHi + 1) * ldk + m0 + lane16];
  b0.x = Bb[(2 * laneHi) * ldb + lane16];
  b0.y = Bb[(2 * laneHi + 1) * ldb + lane16];
#pragma unroll
  for (int kb = 0; kb < CHUNK; kb += 4) {
    v2f a1 = a0, b1 = b0;
    if (kb + 4 < CHUNK) {
      a1.x = Kt[((kb + 4) + 2 * laneHi) * ldk + m0 + lane16];
      a1.y = Kt[((kb + 4) + 2 * laneHi + 1) * ldk + m0 + lane16];
      b1.x = Bb[((kb + 4) + 2 * laneHi) * ldb + lane16];
      b1.y = Bb[((kb + 4) + 2 * laneHi + 1) * ldb + lane16];
    }
    acc = WMMA4(a0, b0, acc);
    a0 = a1; b0 = b1;
  }
  return acc;
}

__global__ __launch_bounds__(256, 1) void e90_dualrate_kernel(
    const float* __restrict__ g_kf, const float* __restrict__ g_vf,
    const float* __restrict__ g_qf, const float* __restrict__ g_df,
    const float* __restrict__ g_ks, const float* __restrict__ g_vs,
    const float* __restrict__ g_qs, const float* __restrict__ g_ds,
    const float* __restrict__ g_g,  const float* __restrict__ g_mf,
    const float* __restrict__ g_ms, const float* __restrict__ g_Sf0,
    const float* __restrict__ g_Ss0,
    float* __restrict__ o_Sf, float* __restrict__ o_Ss, float* __restrict__ o_out)
{
  // resident states
  __shared__ float sSf[KF][VF + 1];
  __shared__ float sSs[KS][VS + 1];
  // double-buffered TDM staging (row pad +1 DWORD inserted by TDM pad engine)
  __shared__ float bQf[2][CHUNK][KF + 1];
  __shared__ float bKf[2][CHUNK][KF + 1];
  __shared__ float bVf[2][CHUNK][VF + 1];
  __shared__ float bQs[2][CHUNK][KS + 1];
  __shared__ float bKs[2][CHUNK][KS + 1];
  __shared__ float bVs[2][CHUNK][VS + 1];
  __shared__ float sAf[CHUNK][17];
  __shared__ float sAs[CHUNK][17];
  __shared__ float sDf[CHUNK], sDs[CHUNK], sG[CHUNK];
  __shared__ float sPf[CHUNK], sPs[CHUNK], sPfi[CHUNK], sPsi[CHUNK];
  __shared__ float sMf[CHUNK], sMs[CHUNK];

  const int bh = blockIdx.x;             // one workgroup per (b,h)
  const int b = bh / H_DIM;
  const int h = bh % H_DIM;
  const int tid = threadIdx.x;
  const int wave = tid >> 5;             // wave32
  const int lane = tid & 31;
  const int lane16 = lane & 15;
  const int laneHi = lane >> 4;

  for (int i = tid; i < KF * VF; i += 256) sSf[i / VF][i % VF] = g_Sf0[bh * KF * VF + i];
  for (int i = tid; i < KS * VS; i += 256) sSs[i / VS][i % VS] = g_Ss0[bh * KS * VS + i];

  // issue TDM loads for chunk 0 into buffer 0 (wave-uniform; EXEC ignored by TDM)
  if (wave == 0) {
    const int base = (0 * B_DIM + b) * H_DIM + h;
    tdm_load_2d((unsigned)(size_t)&bKf[0][0][0], &g_kf[base * KF], KF, CHUNK,
                (unsigned long long)B_DIM * H_DIM * KF, 4);
    tdm_load_2d((unsigned)(size_t)&bQf[0][0][0], &g_qf[base * KF], KF, CHUNK,
                (unsigned long long)B_DIM * H_DIM * KF, 4);
    tdm_load_2d((unsigned)(size_t)&bVf[0][0][0], &g_vf[base * VF], VF, CHUNK,
                (unsigned long long)B_DIM * H_DIM * VF, 5);
    tdm_load_2d((unsigned)(size_t)&bKs[0][0][0], &g_ks[base * KS], KS, CHUNK,
                (unsigned long long)B_DIM * H_DIM * KS, 6);
    tdm_load_2d((unsigned)(size_t)&bQs[0][0][0], &g_qs[base * KS], KS, CHUNK,
                (unsigned long long)B_DIM * H_DIM * KS, 6);
    tdm_load_2d((unsigned)(size_t)&bVs[0][0][0], &g_vs[base * VS], VS, CHUNK,
                (unsigned long long)B_DIM * H_DIM * VS, 5);
  }
  __syncthreads();

  for (int chunk = 0; chunk < NCHUNK; ++chunk) {
    const int t0 = chunk * CHUNK;
    const int cur = chunk & 1;

    // ---- per-timestep scalars + effective slow decay ----
    if (tid < CHUNK) {
      int off = ((t0 + tid) * B_DIM + b) * H_DIM + h;
      float gg = g_g[off];
      sDf[tid] = g_df[off];
      sG[tid]  = gg;
      sDs[tid] = gg * g_ds[off] + (1.0f - gg);   // gated update folds to scalar decay
      sMf[tid] = g_mf[off];
      sMs[tid] = g_ms[off];
    }
    __syncthreads();
    if (tid < CHUNK) {                            // inclusive decay prefix products
      float pf = 1.0f, ps = 1.0f;
      for (int i = 0; i <= tid; ++i) { pf *= sDf[i]; ps *= sDs[i]; }
      sPf[tid] = pf;  sPs[tid] = ps;
      sPfi[tid] = 1.0f / pf;  sPsi[tid] = 1.0f / ps;
    }
    if (wave == 0) __builtin_amdgcn_s_wait_tensorcnt(0);  // chunk's TDM tiles landed
    __syncthreads();

    // ---- wave0: kick off TDM for next chunk into the other buffer ----
    if (wave == 0 && chunk + 1 < NCHUNK) {
      const int nb = cur ^ 1;
      const int base = ((t0 + CHUNK) * B_DIM + b) * H_DIM + h;
      tdm_load_2d((unsigned)(size_t)&bKf[nb][0][0], &g_kf[base * KF], KF, CHUNK,
                  (unsigned long long)B_DIM * H_DIM * KF, 4);
      tdm_load_2d((unsigned)(size_t)&bQf[nb][0][0], &g_qf[base * KF], KF, CHUNK,
                  (unsigned long long)B_DIM * H_DIM * KF, 4);
      tdm_load_2d((unsigned)(size_t)&bVf[nb][0][0], &g_vf[base * VF], VF, CHUNK,
                  (unsigned long long)B_DIM * H_DIM * VF, 5);
      tdm_load_2d((unsigned)(size_t)&bKs[nb][0][0], &g_ks[base * KS], KS, CHUNK,
                  (unsigned long long)B_DIM * H_DIM * KS, 6);
      tdm_load_2d((unsigned)(size_t)&bQs[nb][0][0], &g_qs[base * KS], KS, CHUNK,
                  (unsigned long long)B_DIM * H_DIM * KS, 6);
      tdm_load_2d((unsigned)(size_t)&bVs[nb][0][0], &g_vs[base * VS], VS, CHUNK,
                  (unsigned long long)B_DIM * H_DIM * VS, 5);
    }

    // ---- in-place decay scaling: Q~ = P*q, K~ = k/P (slow: *g) ----
    for (int i = tid; i < CHUNK * KF; i += 256) {
      int t = i / KF, d = i % KF;
      bKf[cur][t][d] *= sPfi[t];
      bQf[cur][t][d] *= sPf[t];
    }
    for (int i = tid; i < CHUNK * KS; i += 256) {
      int t = i / KS, d = i % KS;
      bKs[cur][t][d] *= sG[t] * sPsi[t];
      bQs[cur][t][d] *= sPs[t];
    }
    __syncthreads();

    // ---- attention tiles A = tril(Q~ K~^T); static trip counts per wave ----
    if (wave == 0) {
      v8f acc = {};
      acc = gemm_rt<KF>(&bQf[cur][0][0], KF + 1, &bKf[cur][0][0], KF + 1, acc, lane16, laneHi);
#pragma unroll
      for (int r = 0; r < 8; ++r) {
        int row = r + 8 * laneHi;
        sAf[row][lane16] = (lane16 <= row) ? acc[r] : 0.0f;  // causal incl. diagonal
      }
    } else if (wave == 1) {
      v8f acc = {};
      acc = gemm_rt<KS>(&bQs[cur][0][0], KS + 1, &bKs[cur][0][0], KS + 1, acc, lane16, laneHi);
#pragma unroll
      for (int r = 0; r < 8; ++r) {
        int row = r + 8 * laneHi;
        sAs[row][lane16] = (lane16 <= row) ? acc[r] : 0.0f;
      }
    }
    __syncthreads();

    // ---- main GEMMs: waves 0-3 output tiles, waves 4-7 state tiles ----
    v8f accF = {}, accS = {};
    v8f st[10];
    if (wave < 4) {
      const int n0 = 16 * wave;
      accF = gemm_rr<KF>(&bQf[cur][0][0], KF + 1, &sSf[0][n0], VF + 1, accF, lane16, laneHi);
      accF = gemm_rr<CHUNK>(&sAf[0][0], 17, &bVf[cur][0][n0], VF + 1, accF, lane16, laneHi);
      accS = gemm_rr<KS>(&bQs[cur][0][0], KS + 1, &sSs[0][n0], VS + 1, accS, lane16, laneHi);
      accS = gemm_rr<CHUNK>(&sAs[0][0], 17, &bVs[cur][0][n0], VS + 1, accS, lane16, laneHi);
    } else {
      // 40 state tiles (8 fast + 32 slow): S_new = Pc*(S0 + K~^T @ V)
#pragma unroll
      for (int j = 0; j < 10; ++j) {
        const int idx = (wave - 4) + 4 * j;
        const bool isFast = (idx < 8);
        const int lidx = isFast ? idx : (idx - 8);
        const int mt = lidx >> 2, nt = lidx & 3;
        const float* Sb  = isFast ? &sSf[0][0] : &sSs[0][0];
        const int   ldS  = isFast ? (VF + 1) : (VS + 1);
        const float* Kst = isFast ? &bKf[cur][0][0] : &bKs[cur][0][0];
        const int   ldk  = isFast ? (KF + 1) : (KS + 1);
        const float* Vst = isFast ? &bVf[cur][0][0] : &bVs[cur][0][0];
        const int   ldv  = isFast ? (VF + 1) : (VS + 1);
        v8f c;
#pragma unroll
        for (int r = 0; r < 8; ++r)
          c[r] = Sb[(mt * 16 + r + 8 * laneHi) * ldS + nt * 16 + lane16];
        c = gemm_tr(Kst, ldk, mt * 16, Vst + nt * 16, ldv, c, lane16, laneHi);
        st[j] = c;
      }
    }
    __syncthreads();   // all S0 readers done before state writeback

    if (wave < 4) {
      const int n0 = 16 * wave;
#pragma unroll
      for (int r = 0; r < 8; ++r) {
        int trow = r + 8 * laneHi;
        float o = sMf[trow] * accF[r] + sMs[trow] * accS[r];
        int off = (((t0 + trow) * B_DIM + b) * H_DIM + h) * VF + n0 + lane16;
        o_out[off] = o;
      }
    } else {
      const float PcF = sPf[CHUNK - 1];
      const float PcS = sPs[CHUNK - 1];
#pragma unroll
      for (int j = 0; j < 10; ++j) {
        const int idx = (wave - 4) + 4 * j;
        const bool isFast = (idx < 8);
        const int lidx = isFast ? idx : (idx - 8);
        const int mt = lidx >> 2, nt = lidx & 3;
        float* Sb = isFast ? &sSf[0][0] : &sSs[0][0];
        const int ldS = isFast ? (VF + 1) : (VS + 1);
        const float Pc = isFast ? PcF : PcS;
#pragma unroll
        for (int r = 0; r < 8; ++r)
          Sb[(mt * 16 + r + 8 * laneHi) * ldS + nt * 16 + lane16] = Pc * st[j][r];
      }
    }
    __syncthreads();
  }

  // final states -> global (tuple order: S_f, S_s, output)
  for (int i = tid; i < KF * VF; i += 256) o_Sf[bh * KF * VF + i] = sSf[i / VF][i % VF];
  for (int i = tid; i < KS * VS; i += 256) o_Ss[bh * KS * VS + i] = sSs[i / VS][i % VS];
}

extern "C" void kernel_launch(void* const* d_in, const int* in_sizes, int n_in,
                              void* d_out, int out_size, void* d_ws, size_t ws_size,
                              hipStream_t stream) {
  (void)in_sizes; (void)n_in; (void)out_size; (void)d_ws; (void)ws_size;
  const float* kf  = (const float*)d_in[0];
  const float* vf  = (const float*)d_in[1];
  const float* qf  = (const float*)d_in[2];
  const float* df  = (const float*)d_in[3];
  const float* ks  = (const float*)d_in[4];
  const float* vs  = (const float*)d_in[5];
  const float* qs  = (const float*)d_in[6];
  const float* ds  = (const float*)d_in[7];
  const float* gg  = (const float*)d_in[8];
  const float* mf  = (const float*)d_in[9];
  const float* ms  = (const float*)d_in[10];
  const float* Sf0 = (const float*)d_in[11];
  const float* Ss0 = (const float*)d_in[12];
  float* out  = (float*)d_out;
  float* oSf  = out;                                   // [B,H,32,64]
  float* oSs  = oSf + B_DIM * H_DIM * KF * VF;         // [B,H,128,64]
  float* oOut = oSs + B_DIM * H_DIM * KS * VS;         // [T,B,H,64]
  e90_dualrate_kernel<<<dim3(B_DIM * H_DIM), dim3(256), 0, stream>>>(
      kf, vf, qf, df, ks, vs, qs, ds, gg, mf, ms, Sf0, Ss0, oSf, oSs, oOut);
}